// iSogCLR_Plus_Loss_22643067584622
// MI455X (gfx1250) — compile-verified
//
#include <hip/hip_runtime.h>

typedef __attribute__((ext_vector_type(2))) float v2f;
typedef __attribute__((ext_vector_type(8))) float v8f;

#define ALPHA_W   0.5f
#define GAMMA_S   0.8f
#define GAMMA_U   0.9f
#define RHO_C     6.0f
#define EPS_C     1e-14f
#define ETA_C     0.01f
#define TAU_MIN_C 0.001f
#define TAU_MAX_C 1.0f
#define GCLIP_C   5.0f

__device__ __forceinline__ float waveReduceSum(float v) {
#pragma unroll
  for (int off = 16; off > 0; off >>= 1) v += __shfl_xor(v, off, 32);
  return v;
}
__device__ __forceinline__ float waveReduceMax(float v) {
#pragma unroll
  for (int off = 16; off > 0; off >>= 1) v = fmaxf(v, __shfl_xor(v, off, 32));
  return v;
}

// ---------------- normalize: one wave per row; blockIdx.y selects zis/zjs ----
__global__ void normalize_kernel(const float* __restrict__ zis,
                                 const float* __restrict__ zjs,
                                 float* __restrict__ zin,
                                 float* __restrict__ zjn, int D) {
  int lane = threadIdx.x & 31;
  int w = threadIdx.x >> 5;
  int row = blockIdx.x * 8 + w;
  const float* src = (blockIdx.y == 0 ? zis : zjs) + (size_t)row * D;
  float* dst = (blockIdx.y == 0 ? zin : zjn) + (size_t)row * D;
  float ss = 0.0f;
  for (int k = lane; k < D; k += 32) { float v = src[k]; ss += v * v; }
  ss = waveReduceSum(ss);
  float inv = 1.0f / sqrtf(ss);
  for (int k = lane; k < D; k += 32) dst[k] = src[k] * inv;
}

// ---------------- diag[i] = dot(zin[i], zjn[i]) : one wave per row ----------
__global__ void diag_kernel(const float* __restrict__ zin,
                            const float* __restrict__ zjn,
                            float* __restrict__ diag, int D) {
  int lane = threadIdx.x & 31;
  int w = threadIdx.x >> 5;
  int row = blockIdx.x * 8 + w;
  const float* a = zin + (size_t)row * D;
  const float* b = zjn + (size_t)row * D;
  float s = 0.0f;
  for (int k = lane; k < D; k += 32) s += a[k] * b[k];
  s = waveReduceSum(s);
  if (lane == 0) diag[row] = s;
}

// ---------------- sim = zin * zjn^T via FP32 WMMA 16x16x4 -------------------
// One wave computes a 16x64 strip (four 16x16 N-tiles). Software-pipelined:
// the next K-step's 5 operand loads are issued before the current step's 4
// WMMAs so L2 latency overlaps the matrix pipe instead of serializing it.
// f32 A layout (ISA 7.12.2): lane&15 = M row; VGPR v, half h=lane>>4 -> K=h*2+v.
// f32 B layout mirrors with lane&15 = N col. C/D: VGPR r, M = r + 8*h, N = lane&15.
__global__ void gemm_sim_wmma(const float* __restrict__ A,
                              const float* __restrict__ Bm,
                              float* __restrict__ sim, int Bn, int D) {
  int lane = threadIdx.x & 31;
  int wid = threadIdx.x >> 5;                 // 4 waves per block
  int tm = blockIdx.y;                        // M tile (16 rows)
  int tn0 = (blockIdx.x * 4 + wid) * 4;       // four adjacent N tiles
  int mr = lane & 15;
  int kh = (lane >> 4) * 2;

  const float* aptr = A  + (size_t)(tm * 16 + mr) * D + kh;
  const float* bp0  = Bm + (size_t)((tn0 + 0) * 16 + mr) * D + kh;
  const float* bp1  = Bm + (size_t)((tn0 + 1) * 16 + mr) * D + kh;
  const float* bp2  = Bm + (size_t)((tn0 + 2) * 16 + mr) * D + kh;
  const float* bp3  = Bm + (size_t)((tn0 + 3) * 16 + mr) * D + kh;

  v2f a  = *(const v2f*)(aptr);
  v2f b0 = *(const v2f*)(bp0);
  v2f b1 = *(const v2f*)(bp1);
  v2f b2 = *(const v2f*)(bp2);
  v2f b3 = *(const v2f*)(bp3);

  v8f c0 = {}; v8f c1 = {}; v8f c2 = {}; v8f c3 = {};
#pragma unroll 4
  for (int k0 = 4; k0 < D; k0 += 4) {
    // prefetch next K-step operands (distinct values -> loads stay in flight)
    v2f an  = *(const v2f*)(aptr + k0);
    v2f bn0 = *(const v2f*)(bp0 + k0);
    v2f bn1 = *(const v2f*)(bp1 + k0);
    v2f bn2 = *(const v2f*)(bp2 + k0);
    v2f bn3 = *(const v2f*)(bp3 + k0);
    c0 = __builtin_amdgcn_wmma_f32_16x16x4_f32(false, a, false, b0, (short)0, c0,
                                               false, false);
    c1 = __builtin_amdgcn_wmma_f32_16x16x4_f32(false, a, false, b1, (short)0, c1,
                                               false, false);
    c2 = __builtin_amdgcn_wmma_f32_16x16x4_f32(false, a, false, b2, (short)0, c2,
                                               false, false);
    c3 = __builtin_amdgcn_wmma_f32_16x16x4_f32(false, a, false, b3, (short)0, c3,
                                               false, false);
    a = an; b0 = bn0; b1 = bn1; b2 = bn2; b3 = bn3;
  }
  // drain last K-step
  c0 = __builtin_amdgcn_wmma_f32_16x16x4_f32(false, a, false, b0, (short)0, c0,
                                             false, false);
  c1 = __builtin_amdgcn_wmma_f32_16x16x4_f32(false, a, false, b1, (short)0, c1,
                                             false, false);
  c2 = __builtin_amdgcn_wmma_f32_16x16x4_f32(false, a, false, b2, (short)0, c2,
                                             false, false);
  c3 = __builtin_amdgcn_wmma_f32_16x16x4_f32(false, a, false, b3, (short)0, c3,
                                             false, false);

  int row = tm * 16 + (lane >> 4) * 8;
  int col = lane & 15;
  float* d = sim + (size_t)row * Bn + (size_t)tn0 * 16 + col;
#pragma unroll
  for (int r = 0; r < 8; ++r) {
    size_t o = (size_t)r * Bn;
    d[o]      = c0[r];
    d[o + 16] = c1[r];
    d[o + 32] = c2[r];
    d[o + 48] = c3[r];
  }
}

// ---------------- fused 8-array grid-stride float4 copy ---------------------
struct PtrPack {
  const float* src[8];
  float* dst[8];
};

__global__ void copy8_kernel(PtrPack p, int n) {
  int a = blockIdx.y;
  const float* __restrict__ src = p.src[a];
  float* __restrict__ dst = p.dst[a];
  int n4 = n >> 2;
  int idx = blockIdx.x * blockDim.x + threadIdx.x;
  int stride = gridDim.x * blockDim.x;
  const float4* s4 = (const float4*)src;
  float4* d4 = (float4*)dst;
  for (int i = idx; i < n4; i += stride) d4[i] = s4[i];
  if (idx == 0)
    for (int i = n4 << 2; i < n; ++i) dst[i] = src[i];
}

// ---------------- row (image-side) reductions + state update ----------------
__global__ void row_update_kernel(const float* __restrict__ sim,
                                  const float* __restrict__ diag,
                                  const int* __restrict__ ids,
                                  const float* __restrict__ s_I,
                                  const float* __restrict__ b_I,
                                  const float* __restrict__ u_I,
                                  const float* __restrict__ tau_I,
                                  float* __restrict__ out_g,
                                  float* __restrict__ out_grad,
                                  float* __restrict__ o_s, float* __restrict__ o_b,
                                  float* __restrict__ o_u, float* __restrict__ o_tau,
                                  float* __restrict__ loss_part, int Bn) {
  int i = blockIdx.x;
  int tid = threadIdx.x;
  int lane = tid & 31, wid = tid >> 5;
  __shared__ float redA[8];
  __shared__ float redB[8];
  __shared__ float bc[2];
  int id = ids[i];
  float tau = tau_I[id];
  float invt = 1.0f / tau;
  float dg = diag[i];
  const float* row = sim + (size_t)i * Bn;

  float m = -3.0e38f;
  for (int j = tid; j < Bn; j += 256) m = fmaxf(m, (row[j] - dg) * invt);
  m = waveReduceMax(m);
  if (lane == 0) redA[wid] = m;
  __syncthreads();
  if (tid == 0) {
    float mm = redA[0];
#pragma unroll
    for (int w = 1; w < 8; ++w) mm = fmaxf(mm, redA[w]);
    float ob = b_I[id];
    bc[0] = fmaxf(ob, mm);
    bc[1] = ob;
  }
  __syncthreads();
  float nb = bc[0];

  float g = 0.0f, h = 0.0f;
  for (int j = tid; j < Bn; j += 256) {
    if (j == i) continue;
    float idt = (row[j] - dg) * invt;
    float e = __expf(idt - nb);
    g += e; h += e * idt;
  }
  g = waveReduceSum(g);
  h = waveReduceSum(h);
  if (lane == 0) { redA[wid] = g; redB[wid] = h; }
  __syncthreads();
  if (tid == 0) {
    float G = 0.0f, H = 0.0f;
#pragma unroll
    for (int w = 0; w < 8; ++w) { G += redA[w]; H += redB[w]; }
    float ob = bc[1];
    float s_new = (1.0f - GAMMA_S) * s_I[id] * __expf(ob - nb) + GAMMA_S * G;
    float s_c = fmaxf(s_new, EPS_C);
    float wsum = H / s_c;
    float grad = __logf(s_c) + nb + RHO_C - wsum / (float)(Bn - 1);
    float gc = fminf(fmaxf(grad, -GCLIP_C), GCLIP_C);
    float u_new = (1.0f - GAMMA_U) * u_I[id] + GAMMA_U * gc;
    float tau_new = fminf(fmaxf(tau - ETA_C * u_new, TAU_MIN_C), TAU_MAX_C);
    out_g[i] = G;
    out_grad[i] = grad;
    o_s[id] = s_new; o_b[id] = nb; o_u[id] = u_new; o_tau[id] = tau_new;
    loss_part[i] = tau * wsum;       // image_loss_i
  }
}

// ---------------- column (text-side) reductions + state update --------------
// Block = 8 waves; lane -> column (coalesced reads), waves stride over rows.
__global__ void col_update_kernel(const float* __restrict__ sim,
                                  const float* __restrict__ diag,
                                  const int* __restrict__ ids,
                                  const float* __restrict__ s_T,
                                  const float* __restrict__ b_T,
                                  const float* __restrict__ u_T,
                                  const float* __restrict__ tau_T,
                                  float* __restrict__ out_g,
                                  float* __restrict__ out_grad,
                                  float* __restrict__ o_s, float* __restrict__ o_b,
                                  float* __restrict__ o_u, float* __restrict__ o_tau,
                                  float* __restrict__ loss_part, int Bn) {
  int tid = threadIdx.x;
  int lane = tid & 31, w = tid >> 5;
  int c = blockIdx.x * 32 + lane;
  __shared__ float sm[8][32];
  __shared__ float sg[8][32];
  __shared__ float sh[8][32];
  __shared__ float snb[32];
  int id = ids[c];
  float tau = tau_T[id];
  float invt = 1.0f / tau;
  float dg = diag[c];

  float m = -3.0e38f;
  for (int i = w; i < Bn; i += 8)
    m = fmaxf(m, (sim[(size_t)i * Bn + c] - dg) * invt);
  sm[w][lane] = m;
  __syncthreads();
  if (tid < 32) {
    float mm = sm[0][tid];
#pragma unroll
    for (int q = 1; q < 8; ++q) mm = fmaxf(mm, sm[q][tid]);
    snb[tid] = fmaxf(b_T[id], mm);
  }
  __syncthreads();
  float nb = snb[lane];

  float g = 0.0f, h = 0.0f;
  for (int i = w; i < Bn; i += 8) {
    if (i == c) continue;
    float tdt = (sim[(size_t)i * Bn + c] - dg) * invt;
    float e = __expf(tdt - nb);
    g += e; h += e * tdt;
  }
  sg[w][lane] = g; sh[w][lane] = h;
  __syncthreads();
  if (tid < 32) {
    float G = 0.0f, H = 0.0f;
#pragma unroll
    for (int q = 0; q < 8; ++q) { G += sg[q][tid]; H += sh[q][tid]; }
    float ob = b_T[id];
    float nb0 = snb[tid];
    float s_new = (1.0f - GAMMA_S) * s_T[id] * __expf(ob - nb0) + GAMMA_S * G;
    float s_c = fmaxf(s_new, EPS_C);
    float wsum = H / s_c;
    float grad = __logf(s_c) + nb0 + RHO_C - wsum / (float)(Bn - 1);
    float gc = fminf(fmaxf(grad, -GCLIP_C), GCLIP_C);
    float u_new = (1.0f - GAMMA_U) * u_T[id] + GAMMA_U * gc;
    float tau_new = fminf(fmaxf(tau - ETA_C * u_new, TAU_MIN_C), TAU_MAX_C);
    out_g[c] = G;
    out_grad[c] = grad;
    o_s[id] = s_new; o_b[id] = nb0; o_u[id] = u_new; o_tau[id] = tau_new;
    loss_part[c] = tau * wsum;       // text_loss_j
  }
}

// ---------------- deterministic single-block finalize -----------------------
__global__ void finalize_kernel(const float* __restrict__ wsimg,
                                const float* __restrict__ wstxt,
                                const int* __restrict__ ids,
                                const float* __restrict__ tau_I,
                                const float* __restrict__ tau_T,
                                float* __restrict__ out, int Bn) {
  __shared__ float r[4][256];
  int tid = threadIdx.x;
  float a = 0.0f, b = 0.0f, c = 0.0f, d = 0.0f;
  for (int i = tid; i < Bn; i += 256) {
    a += wsimg[i]; b += wstxt[i];
    int id = ids[i];
    c += tau_I[id]; d += tau_T[id];
  }
  r[0][tid] = a; r[1][tid] = b; r[2][tid] = c; r[3][tid] = d;
  __syncthreads();
  for (int s = 128; s > 0; s >>= 1) {
    if (tid < s) {
#pragma unroll
      for (int q = 0; q < 4; ++q) r[q][tid] += r[q][tid + s];
    }
    __syncthreads();
  }
  if (tid == 0) {
    float invB = 1.0f / (float)Bn;
    out[4 * Bn]     = ALPHA_W * r[0][0] * invB + (1.0f - ALPHA_W) * r[1][0] * invB;
    out[4 * Bn + 1] = r[2][0] * invB;
    out[4 * Bn + 2] = r[3][0] * invB;
  }
}

extern "C" void kernel_launch(void* const* d_in, const int* in_sizes, int n_in,
                              void* d_out, int out_size, void* d_ws, size_t ws_size,
                              hipStream_t stream) {
  const float* zis   = (const float*)d_in[0];
  const float* zjs   = (const float*)d_in[1];
  const int*   ids   = (const int*)d_in[2];
  const float* s_I   = (const float*)d_in[3];
  const float* s_T   = (const float*)d_in[4];
  const float* b_I   = (const float*)d_in[5];
  const float* b_T   = (const float*)d_in[6];
  const float* u_I   = (const float*)d_in[7];
  const float* u_T   = (const float*)d_in[8];
  const float* tau_I = (const float*)d_in[9];
  const float* tau_T = (const float*)d_in[10];

  const int Bn = in_sizes[2];          // 4096
  const int D  = in_sizes[0] / Bn;     // 256
  const int N  = in_sizes[3];          // 9000000

  float* out = (float*)d_out;
  float* ws  = (float*)d_ws;

  // workspace layout (floats)
  float* zin  = ws;
  float* zjn  = zin + (size_t)Bn * D;
  float* diag = zjn + (size_t)Bn * D;
  float* wimg = diag + Bn;
  float* wtxt = wimg + Bn;
  size_t simOff = (((size_t)(wtxt + Bn - ws) + 255) / 256) * 256;
  float* sim  = ws + simOff;

  // output layout
  float* out_gi  = out;
  float* out_gt  = out + Bn;
  float* out_gti = out + 2 * (size_t)Bn;
  float* out_gtt = out + 3 * (size_t)Bn;
  size_t base = 4 * (size_t)Bn + 3;
  float* o_sI = out + base;
  float* o_sT = o_sI + N;
  float* o_bI = o_sT + N;
  float* o_bT = o_bI + N;
  float* o_uI = o_bT + N;
  float* o_uT = o_uI + N;
  float* o_tauI = o_uT + N;
  float* o_tauT = o_tauI + N;

  // 1) normalize both embedding matrices
  normalize_kernel<<<dim3(Bn / 8, 2), 256, 0, stream>>>(zis, zjs, zin, zjn, D);
  // 2) diagonal similarities
  diag_kernel<<<Bn / 8, 256, 0, stream>>>(zin, zjn, diag, D);
  // 3) FP32 WMMA GEMM: sim = zin * zjn^T  (wave = 16x64 strip, pipelined)
  gemm_sim_wmma<<<dim3(Bn / 256, Bn / 16), 128, 0, stream>>>(zin, zjn, sim, Bn, D);
  // 4) state passthrough copies (bandwidth floor: 8 x 36 MB each way), fused
  {
    PtrPack p;
    p.src[0] = s_I;   p.dst[0] = o_sI;
    p.src[1] = s_T;   p.dst[1] = o_sT;
    p.src[2] = b_I;   p.dst[2] = o_bI;
    p.src[3] = b_T;   p.dst[3] = o_bT;
    p.src[4] = u_I;   p.dst[4] = o_uI;
    p.src[5] = u_T;   p.dst[5] = o_uT;
    p.src[6] = tau_I; p.dst[6] = o_tauI;
    p.src[7] = tau_T; p.dst[7] = o_tauT;
    int n4 = N >> 2;
    int grid = (n4 + 255) / 256;
    copy8_kernel<<<dim3(grid, 8), 256, 0, stream>>>(p, N);
  }
  // 5) row / column reductions + scatter updates into the copies
  row_update_kernel<<<Bn, 256, 0, stream>>>(sim, diag, ids, s_I, b_I, u_I, tau_I,
                                            out_gi, out_gti, o_sI, o_bI, o_uI,
                                            o_tauI, wimg, Bn);
  col_update_kernel<<<Bn / 32, 256, 0, stream>>>(sim, diag, ids, s_T, b_T, u_T,
                                                 tau_T, out_gt, out_gtt, o_sT,
                                                 o_bT, o_uT, o_tauT, wtxt, Bn);
  // 6) deterministic scalar outputs
  finalize_kernel<<<1, 256, 0, stream>>>(wimg, wtxt, ids, tau_I, tau_T, out, Bn);
}